// SpatialCorrelationSampler_75282186764380
// MI455X (gfx1250) — compile-verified
//
#include <hip/hip_runtime.h>

// ---------------------------------------------------------------------------
// Spatial correlation sampler, MI455X (gfx1250), fp32.
//
// Stage 1 (corr_dot_wmma): D[b,y,x] = sum_c in1[b,c,y,x]*in2[b,c,y,x]
//   Bandwidth-bound: 151 MB streamed reads. Channel reduction is done with
//   V_WMMA_F32_16X16X4_F32 using an all-ones A matrix:
//       C[m][n] += sum_k 1 * P[k][n],  P = in1*in2
//   Every row of C equals D for the wave's 16 pixels. Since A is all ones the
//   result is invariant to the K striping of the f32 B layout; only the
//   (documented) N=lane column striping matters.
//
// Stage 2 (corr_gather): out[b, py*21+px, y, x] = Dpad[b, y+(py-10)*2, x+(px-10)*2]
//   130 MB of NT vector stores; D (295 KB) stays resident in L2.
// ---------------------------------------------------------------------------

typedef __attribute__((ext_vector_type(2))) float v2f;
typedef __attribute__((ext_vector_type(4))) float v4f;
typedef __attribute__((ext_vector_type(8))) float v8f;

#define BATCH   8
#define CHAN    256
#define HH      96
#define WW      96
#define HWSZ    (HH * WW)            // 9216
#define PATCH   21
#define NPATCH  (PATCH * PATCH)      // 441
#define HALFP   10
#define DILP    2

// ---------------- Stage 1: channel dot product via WMMA --------------------
// grid: 576 blocks * 256 threads; wave handles 16 consecutive linear pixels.
// 9216 % 16 == 0 so a wave never straddles a batch image.
__global__ __launch_bounds__(256) void corr_dot_wmma(
    const float* __restrict__ in1,
    const float* __restrict__ in2,
    float* __restrict__ D)
{
    const int lane   = threadIdx.x & 31;
    const int waveG  = blockIdx.x * 8 + (threadIdx.x >> 5);
    const int pixB   = waveG * 16;              // first pixel of this wave
    const int p      = lane & 15;               // pixel column n = lane & 15
    const int pix    = pixB + p;
    const int b      = pix / HWSZ;
    const int rem    = pix - b * HWSZ;
    const int koff   = (lane >> 4) * 2;         // lanes 0-15: ch c,c+1 ; lanes 16-31: c+2,c+3
    const int base   = b * (CHAN * HWSZ) + rem;

    v2f aOnes;                                  // all-ones A matrix (16x4 f32 = 2 VGPRs)
    aOnes.x = 1.0f;
    aOnes.y = 1.0f;
    v8f acc = {};                               // 16x16 f32 accumulator (8 VGPRs)

#pragma unroll 8
    for (int c = 0; c < CHAN; c += 4) {
        const int i0 = base + (c + koff) * HWSZ;
        const int i1 = i0 + HWSZ;
        // streamed once -> non-temporal loads, keep L2 for D reuse in stage 2
        float x0 = __builtin_nontemporal_load(in1 + i0);
        float x1 = __builtin_nontemporal_load(in1 + i1);
        float y0 = __builtin_nontemporal_load(in2 + i0);
        float y1 = __builtin_nontemporal_load(in2 + i1);
        v2f bm;                                 // B matrix (4x16 f32 = 2 VGPRs), N = lane&15
        bm.x = x0 * y0;
        bm.y = x1 * y1;
        acc = __builtin_amdgcn_wmma_f32_16x16x4_f32(
            /*neg_a=*/false, aOnes,
            /*neg_b=*/false, bm,
            /*c_mod=*/(short)0, acc,
            /*reuse_a=*/false, /*reuse_b=*/false);
    }

    // Row 0 of C: VGPR0, lanes 0-15 -> D for the 16 pixels.
    if (lane < 16) D[pix] = acc[0];
}

// ---------------- Stage 2: shifted broadcast into 441 channels -------------
// One thread per 4 consecutive x (16B NT store). D reads served from L2.
__global__ __launch_bounds__(256) void corr_gather(
    const float* __restrict__ D,
    float* __restrict__ out)
{
    const int W4  = WW / 4;                     // 24 quads per row
    const int tid = blockIdx.x * blockDim.x + threadIdx.x;

    int t = tid;
    const int xq = t % W4;  t /= W4;
    const int y  = t % HH;  t /= HH;
    const int pp = t % NPATCH; t /= NPATCH;
    const int b  = t;
    if (b >= BATCH) return;

    const int py = pp / PATCH;
    const int px = pp - py * PATCH;
    const int sy = y + (py - HALFP) * DILP;

    v4f v = {};
    if (sy >= 0 && sy < HH) {
        const float* Drow = D + b * HWSZ + sy * WW;
        const int x0 = xq * 4 + (px - HALFP) * DILP;
#pragma unroll
        for (int i = 0; i < 4; ++i) {
            const int sx = x0 + i;
            v[i] = (sx >= 0 && sx < WW) ? Drow[sx] : 0.0f;
        }
    }
    // streamed output -> non-temporal 16B store, coalesced across the wave
    __builtin_nontemporal_store(v, (v4f*)out + tid);
}

// ---------------------------------------------------------------------------
extern "C" void kernel_launch(void* const* d_in, const int* in_sizes, int n_in,
                              void* d_out, int out_size, void* d_ws, size_t ws_size,
                              hipStream_t stream)
{
    const float* in1 = (const float*)d_in[0];
    const float* in2 = (const float*)d_in[1];
    float*       out = (float*)d_out;
    float*       Dws = (float*)d_ws;            // needs 8*96*96*4 = 294912 bytes

    // Stage 1: 73728 pixels / (8 waves * 16 pixels) = 576 blocks
    corr_dot_wmma<<<576, 256, 0, stream>>>(in1, in2, Dws);

    // Stage 2: 8*441*96*24 = 8,128,512 quads / 256 = 31752 blocks (exact)
    corr_gather<<<31752, 256, 0, stream>>>(Dws, out);
}